// DirectedEdgeMessage_89885075571226
// MI455X (gfx1250) — compile-verified
//
#include <hip/hip_runtime.h>
#include <hip/hip_bf16.h>

// Problem constants (from reference setup_inputs):
//   B=256 batches, A=64 atoms, E=128 edges, K=6 neighbors, H=256 hidden
#define BATCHES 256
#define NATOMS  64
#define NEDGES  128
#define NNBR    6
#define HID     256

// Native clang vector type: accepted by __builtin_nontemporal_store and
// lowers to b128 LDS/global accesses.
typedef __attribute__((ext_vector_type(4))) float v4f;

// Raw LDS byte offset of a generic pointer that points into LDS.
typedef __attribute__((address_space(3))) unsigned char lds_byte_t;
static __device__ __forceinline__ unsigned lds_off(const void* p) {
    return (unsigned)(unsigned long long)(lds_byte_t*)p;
}

__global__ __launch_bounds__(256) void DirectedEdgeMessage_89885075571226_kernel(
    const float* __restrict__ rep,    // [1, B, E, H] f32
    const int*   __restrict__ pairs,  // [B, E, 2]    i32
    const int*   __restrict__ nbrs,   // [B, E, K]    i32
    const float* __restrict__ xyz,    // [B, A, 3]    f32
    float*       __restrict__ out)    // [1, B, E, H] f32
{
    extern __shared__ float smem[];
    float* s_rep  = smem;                       // E*H   = 32768 f32 (131072 B)
    float* s_dist = smem + NEDGES * HID;        // E     = 128 f32
    int*   s_nbr  = (int*)(s_dist + NEDGES);    // E*K   = 768 i32

    const int b   = blockIdx.x;
    const int tid = threadIdx.x;

    // ---------------------------------------------------------------
    // Phase 1: async-stage this batch's edge block rep[b] into LDS.
    // 131072 B = 8192 b128 transfers; 256 lanes x 32 issues each.
    // CDNA5 path: GLOBAL_LOAD_ASYNC_TO_LDS_B128 (ASYNCcnt), GVS mode.
    // INST_OFFSET is added to BOTH the LDS dest and the global source,
    // so one (lds, goff) VGPR pair + literal offsets covers all 32 —
    // zero per-transfer VALU address math. Data is read once -> NT.
    // ---------------------------------------------------------------
    const float* gbase = rep + (size_t)b * (NEDGES * HID);
    const unsigned lds_a = lds_off(s_rep) + (unsigned)tid * 16u;
    const unsigned g_off = (unsigned)tid * 16u;

#define ASYNC_LD(IOFF)                                                        \
    asm volatile("global_load_async_to_lds_b128 %0, %1, %2 offset:" #IOFF     \
                 " th:TH_LOAD_NT"                                             \
                 :: "v"(lds_a), "v"(g_off), "s"(gbase) : "memory")

    ASYNC_LD(0);      ASYNC_LD(4096);   ASYNC_LD(8192);   ASYNC_LD(12288);
    ASYNC_LD(16384);  ASYNC_LD(20480);  ASYNC_LD(24576);  ASYNC_LD(28672);
    ASYNC_LD(32768);  ASYNC_LD(36864);  ASYNC_LD(40960);  ASYNC_LD(45056);
    ASYNC_LD(49152);  ASYNC_LD(53248);  ASYNC_LD(57344);  ASYNC_LD(61440);
    ASYNC_LD(65536);  ASYNC_LD(69632);  ASYNC_LD(73728);  ASYNC_LD(77824);
    ASYNC_LD(81920);  ASYNC_LD(86016);  ASYNC_LD(90112);  ASYNC_LD(94208);
    ASYNC_LD(98304);  ASYNC_LD(102400); ASYNC_LD(106496); ASYNC_LD(110592);
    ASYNC_LD(114688); ASYNC_LD(118784); ASYNC_LD(122880); ASYNC_LD(126976);
#undef ASYNC_LD

    // ---------------------------------------------------------------
    // Phase 2 (overlaps with the in-flight async copies):
    //   (a) stage neighbor indices into LDS
    //   (b) compute reciprocal-squared-distance weights
    // ---------------------------------------------------------------
    for (int i = tid; i < NEDGES * NNBR; i += 256)
        s_nbr[i] = nbrs[(size_t)b * (NEDGES * NNBR) + i];

    if (tid < NEDGES) {
        const int p0 = pairs[((size_t)b * NEDGES + tid) * 2 + 0];
        const int p1 = pairs[((size_t)b * NEDGES + tid) * 2 + 1];
        const float* x0 = xyz + ((size_t)b * NATOMS + p0) * 3;
        const float* x1 = xyz + ((size_t)b * NATOMS + p1) * 3;
        const float dx = x0[0] - x1[0];
        const float dy = x0[1] - x1[1];
        const float dz = x0[2] - x1[2];
        const float d2 = dx * dx + dy * dy + dz * dz;
        float inv = 1.0f / d2;
        if (__builtin_isinf(inv)) inv = 0.0f;   // jnp.where(isinf, 0, inv)
        s_dist[tid] = inv;
    }

    // Per-wave wait for our async transfers, then cross-wave barrier so
    // every thread sees the full LDS-resident edge block.
    asm volatile("s_wait_asynccnt 0" ::: "memory");
    __syncthreads();

    // ---------------------------------------------------------------
    // Phase 3: gather + weighted sum.
    // Flat float4 index f in [0, E*H/4): e = f>>6, quad = f&63.
    // A wave32 covers 32 consecutive quads of one row -> uniform e,
    // conflict-free ds_load_b128, broadcast s_dist reads, coalesced
    // non-temporal b128 stores (output is never re-read).
    // ---------------------------------------------------------------
    float* outb = out + (size_t)b * (NEDGES * HID);
#pragma unroll 2
    for (int i = 0; i < (NEDGES * HID / 4) / 256; ++i) {          // 32 iters
        const int f  = i * 256 + tid;
        const int e  = f >> 6;
        const int q  = f & 63;
        const int* nb = s_nbr + e * NNBR;

        v4f acc = {0.0f, 0.0f, 0.0f, 0.0f};
#pragma unroll
        for (int k = 0; k < NNBR; ++k) {
            const int   n = nb[k];
            const float w = s_dist[n];
            const v4f   r =
                *reinterpret_cast<const v4f*>(s_rep + n * HID + q * 4);
            acc += w * r;   // contracts to v_fma_f32 under HIP default fp-contract
        }
        __builtin_nontemporal_store(
            acc, reinterpret_cast<v4f*>(outb + (size_t)f * 4));
    }
}

extern "C" void kernel_launch(void* const* d_in, const int* in_sizes, int n_in,
                              void* d_out, int out_size, void* d_ws, size_t ws_size,
                              hipStream_t stream) {
    (void)in_sizes; (void)n_in; (void)out_size; (void)d_ws; (void)ws_size;

    const float* rep   = (const float*)d_in[0];  // bond_representations f32
    const int*   pairs = (const int*)  d_in[1];  // bond_pairs i32
    const int*   nbrs  = (const int*)  d_in[2];  // bond_neighbors i32
    const float* xyz   = (const float*)d_in[3];  // xyz f32
    float*       out   = (float*)d_out;

    // LDS: 131072 (rep block) + 512 (dist) + 3072 (neighbors) = 134656 B.
    // gfx1250 allows up to 320 KB per workgroup (2 blocks/WGP resident).
    const size_t smem_bytes =
        (size_t)(NEDGES * HID + NEDGES) * sizeof(float) +
        (size_t)(NEDGES * NNBR) * sizeof(int);

    DirectedEdgeMessage_89885075571226_kernel
        <<<dim3(BATCHES), dim3(256), smem_bytes, stream>>>(rep, pairs, nbrs, xyz, out);
}